// CustomProjectionModel_27745488732738
// MI455X (gfx1250) — compile-verified
//
#include <hip/hip_runtime.h>

#define IN_DIM   512
#define HID_DIM  2048
#define OUT_DIM  512
#define MDIM     64
#define N_W1     (HID_DIM * IN_DIM)
#define N_B1     (HID_DIM)
#define N_W2     (OUT_DIM * HID_DIM)
#define N_B2     (OUT_DIM)
#define TOTAL_PARAMS (N_W1 + N_B1 + N_W2 + N_B2)   // 2,099,712
#define BATCH    256

typedef __attribute__((ext_vector_type(2))) float v2f;
typedef __attribute__((ext_vector_type(8))) float v8f;

// ---------------------------------------------------------------------------
// Stage 1: flat = P @ u.  One wave32 per row of P (64 fp32 = 256B contiguous).
// Lane l loads float2 at column 2l -> fully coalesced 256B per wave.
// Butterfly shuffle reduce, lane 0 writes the scalar result.
// HBM-bound: 537 MB streamed once ~= 23 us at 23.3 TB/s.
// ---------------------------------------------------------------------------
__global__ void __launch_bounds__(256) matvec_Pu_kernel(
    const float* __restrict__ P,
    const float* __restrict__ u,
    float* __restrict__ flat)
{
    const int lane = threadIdx.x & 31;
    const int row  = (blockIdx.x * blockDim.x + threadIdx.x) >> 5;  // grid sized exactly

    const v2f p  = *(const v2f*)(P + (size_t)row * MDIM + 2 * lane);
    const v2f uu = *(const v2f*)(u + 2 * lane);

    float part = p.x * uu.x + p.y * uu.y;
    #pragma unroll
    for (int off = 16; off >= 1; off >>= 1)
        part += __shfl_xor(part, off, 32);

    if (lane == 0)
        flat[row] = part;
}

// ---------------------------------------------------------------------------
// Stages 2/3: C[M x N] = act(A[M x K] @ W[N x K]^T + bias), fp32 WMMA.
// One wave computes a 16(M) x 64(N) strip: 4 accumulators reuse each A tile.
// f32 16x16x4 operand layout (ISA 7.12.2): element (r, k0+2*kh+j) lives in
// lane = (r&15) + 16*kh, vgpr j  -- identical for A (r=M-row) and B (r=N-row
// of the row-major W, i.e. B = W^T), so both load as float2 at row*K + 2*kh.
// ---------------------------------------------------------------------------
__device__ __forceinline__ void store_ctile(
    float* __restrict__ C, const v8f acc, const float* __restrict__ bias,
    int mBase, int col, int N, int kh, int doRelu)
{
    const float bv = bias[col];
    #pragma unroll
    for (int v = 0; v < 8; ++v) {
        float val = acc[v] + bv;
        if (doRelu) val = fmaxf(val, 0.0f);
        C[(size_t)(mBase + v + 8 * kh) * N + col] = val;
    }
}

__global__ void __launch_bounds__(256) gemm_bias_wmma_kernel(
    const float* __restrict__ A,     // M x K, row-major
    const float* __restrict__ W,     // N x K, row-major (used as B = W^T)
    const float* __restrict__ bias,  // N
    float* __restrict__ C,           // M x N, row-major
    int K, int nGroups, int N, int doRelu)
{
    const int lane = threadIdx.x & 31;
    const int wave = (blockIdx.x * blockDim.x + threadIdx.x) >> 5;

    const int mTile  = wave / nGroups;
    const int nGroup = wave % nGroups;
    const int mBase  = mTile * 16;
    const int nBase  = nGroup * 64;

    const int r  = lane & 15;   // M-row (A) / N-row (B) within tile
    const int kh = lane >> 4;   // which K-half-pair

    const float* aPtr = A + (size_t)(mBase + r) * K + 2 * kh;
    const float* wPtr = W + (size_t)(nBase + r) * K + 2 * kh;
    const size_t wStride16 = (size_t)16 * K;

    v8f acc0 = {}, acc1 = {}, acc2 = {}, acc3 = {};

    #pragma unroll 2
    for (int k0 = 0; k0 < K; k0 += 4) {
        const v2f a  = *(const v2f*)(aPtr + k0);
        const v2f b0 = *(const v2f*)(wPtr + k0);
        const v2f b1 = *(const v2f*)(wPtr + wStride16 + k0);
        const v2f b2 = *(const v2f*)(wPtr + 2 * wStride16 + k0);
        const v2f b3 = *(const v2f*)(wPtr + 3 * wStride16 + k0);
        acc0 = __builtin_amdgcn_wmma_f32_16x16x4_f32(false, a, false, b0, (short)0, acc0, false, false);
        acc1 = __builtin_amdgcn_wmma_f32_16x16x4_f32(false, a, false, b1, (short)0, acc1, false, false);
        acc2 = __builtin_amdgcn_wmma_f32_16x16x4_f32(false, a, false, b2, (short)0, acc2, false, false);
        acc3 = __builtin_amdgcn_wmma_f32_16x16x4_f32(false, a, false, b3, (short)0, acc3, false, false);
    }

    store_ctile(C, acc0, bias, mBase, nBase +  0 + r, N, kh, doRelu);
    store_ctile(C, acc1, bias, mBase, nBase + 16 + r, N, kh, doRelu);
    store_ctile(C, acc2, bias, mBase, nBase + 32 + r, N, kh, doRelu);
    store_ctile(C, acc3, bias, mBase, nBase + 48 + r, N, kh, doRelu);
}

// ---------------------------------------------------------------------------
// Launcher
// ---------------------------------------------------------------------------
extern "C" void kernel_launch(void* const* d_in, const int* in_sizes, int n_in,
                              void* d_out, int out_size, void* d_ws, size_t ws_size,
                              hipStream_t stream)
{
    const float* x = (const float*)d_in[0];   // 256 x 512
    const float* P = (const float*)d_in[1];   // TOTAL_PARAMS x 64
    const float* u = (const float*)d_in[2];   // 64

    float* out  = (float*)d_out;              // 256 x 512
    float* flat = (float*)d_ws;               // TOTAL_PARAMS floats (8.4 MB)
    float* h    = flat + TOTAL_PARAMS;        // 256 x 2048 floats (2 MB)

    // Stage 1: flat = P @ u   (one wave per row; 8 rows per 256-thread block)
    matvec_Pu_kernel<<<TOTAL_PARAMS / 8, 256, 0, stream>>>(P, u, flat);

    const float* W1 = flat;
    const float* b1 = flat + N_W1;
    const float* W2 = flat + N_W1 + N_B1;
    const float* b2 = flat + N_W1 + N_B1 + N_W2;

    // Stage 2: h = relu(x @ W1^T + b1)
    // waves = (256/16) * (2048/64) = 512 -> 64 blocks of 8 waves
    gemm_bias_wmma_kernel<<<64, 256, 0, stream>>>(x, W1, b1, h,
                                                  IN_DIM, HID_DIM / 64, HID_DIM, 1);

    // Stage 3: out = h @ W2^T + b2
    // waves = (256/16) * (512/64) = 128 -> 16 blocks of 8 waves
    gemm_bias_wmma_kernel<<<16, 256, 0, stream>>>(h, W2, b2, out,
                                                  HID_DIM, OUT_DIM / 64, OUT_DIM, 0);
}